// FieldModel_75840532513213
// MI455X (gfx1250) — compile-verified
//
#include <hip/hip_runtime.h>

#define BATCH 2
#define NATOM 8192
#define MNB   64
#define DIM   32
#define WPB   4                      // waves per block
#define NBS   12                     // per-neighbor LDS scratch stride (floats)
#define HSTR  33                     // padded h-row stride (bank-conflict-free)
#define LDSW  (MNB*NBS + MNB*HSTR)   // floats of LDS per wave

typedef float v2f __attribute__((ext_vector_type(2)));
typedef float v8f __attribute__((ext_vector_type(8)));

__global__ __launch_bounds__(WPB * 32) void field_fused_kernel(
    const int*   __restrict__ element_map,
    const int*   __restrict__ neighbor_indices,
    const int*   __restrict__ neighbor_types,
    const float* __restrict__ neighbor_vectors,
    const float* __restrict__ type_embed,
    const float* __restrict__ elem_bias,
    const float* __restrict__ W1, const float* __restrict__ b1,
    const float* __restrict__ W2, const float* __restrict__ b2,
    const float* __restrict__ W3, const float* __restrict__ b3,
    float* __restrict__ out_Etot,   // [2]
    float* __restrict__ out_Ei,     // [b*n]
    float* __restrict__ out_force,  // [b*n*3]
    float* __restrict__ out_virial, // [b*9]
    float* __restrict__ out_av)     // [b*n*9]
{
  __shared__ float lds[WPB * LDSW];
  const int lane  = threadIdx.x & 31;
  const int wave  = threadIdx.x >> 5;
  const int atom  = blockIdx.x * WPB + wave;       // 0 .. b*n-1
  const int batch = atom / NATOM;
  const long base = (long)atom * MNB;

  float* nbS = lds + wave * LDSW;                  // [64][12] geometry scratch
  float* hS  = nbS + MNB * NBS;                    // [64][33] hidden activations

  const float PI      = 3.14159265358979323846f;
  const float PI_CUT  = PI / 6.0f;

  // ---------------- Phase 1: per-neighbor geometry & features -> LDS --------
  #pragma unroll
  for (int rep = 0; rep < 2; ++rep) {
    int j = lane + rep * 32;
    const float* nv = neighbor_vectors + (base + j) * 4;
    float rx = nv[1], ry = nv[2], rz = nv[3];
    float d = sqrtf(rx * rx + ry * ry + rz * rz);
    float s = 0.0f, sp = 0.0f;
    if (d < 6.0f) {
      float ph  = PI_CUT * d;
      float fc  = 0.5f * (cosf(ph) + 1.0f);
      float fcp = -0.5f * PI_CUT * sinf(ph);
      float e   = expf(-d);
      s  = e * fc;
      sp = e * (fcp - fc);                          // d(s)/d(d)
    }
    float* row = nbS + j * NBS;
    row[0] = s;       row[1] = s * rx; row[2] = s * ry; row[3] = s * rz;
    row[4] = rx;      row[5] = ry;     row[6] = rz;     row[7] = d;
    row[8] = sp;
    row[9] = (float)neighbor_types[base + j];
  }
  __syncthreads();

  // ---------------- Phase 2: feat(16x4) @ W1(4x32) via f32 WMMA -------------
  const int colLo = lane & 15;
  const int colHi = colLo + 16;
  const int kb    = (lane < 16) ? 0 : 2;           // K rows held by this lane
  v2f Blo, Bhi;
  Blo[0] = W1[kb * DIM + colLo];      Blo[1] = W1[(kb + 1) * DIM + colLo];
  Bhi[0] = W1[kb * DIM + colHi];      Bhi[1] = W1[(kb + 1) * DIM + colHi];
  const float b1lo = b1[colLo], b1hi = b1[colHi];
  const float teL0 = type_embed[0 * DIM + colLo], teL1 = type_embed[1 * DIM + colLo],
              teL2 = type_embed[2 * DIM + colLo];
  const float teH0 = type_embed[0 * DIM + colHi], teH1 = type_embed[1 * DIM + colHi],
              teH2 = type_embed[2 * DIM + colHi];

  const int rowOff = (lane >> 4) << 3;             // C rows: 0 or 8
  float gl = 0.0f, gh = 0.0f;
  #pragma unroll
  for (int t = 0; t < 4; ++t) {
    int jr = t * 16 + colLo;                       // A row = neighbor in tile
    v2f A;
    A[0] = nbS[jr * NBS + kb];
    A[1] = nbS[jr * NBS + kb + 1];
    v8f cz = {0.f, 0.f, 0.f, 0.f, 0.f, 0.f, 0.f, 0.f};
    v8f clo = __builtin_amdgcn_wmma_f32_16x16x4_f32(false, A, false, Blo,
                                                    (short)0, cz, false, false);
    v8f chi = __builtin_amdgcn_wmma_f32_16x16x4_f32(false, A, false, Bhi,
                                                    (short)0, cz, false, false);
    #pragma unroll
    for (int v = 0; v < 8; ++v) {
      int jj = t * 16 + rowOff + v;                // neighbor index of this C row
      int ty = (int)nbS[jj * NBS + 9];
      float teL = (ty == 0) ? teL0 : ((ty == 1) ? teL1 : teL2);
      float teH = (ty == 0) ? teH0 : ((ty == 1) ? teH1 : teH2);
      float hlo = tanhf(clo[v] + b1lo + teL);
      float hhi = tanhf(chi[v] + b1hi + teH);
      hS[jj * HSTR + colLo] = hlo;
      hS[jj * HSTR + colHi] = hhi;
      gl += hlo; gh += hhi;
    }
  }
  __syncthreads();
  gl += __shfl_xor(gl, 16, 32);
  gh += __shfl_xor(gh, 16, 32);
  float g = ((lane < 16) ? gl : gh) * (1.0f / MNB);   // lane L holds g[L]

  // ---------------- Phase 3: g @ W2, Ei ------------------------------------
  float pre2 = b2[lane];
  for (int k = 0; k < DIM; ++k) {
    float gk = __shfl(g, k, 32);
    pre2 = fmaf(gk, W2[k * DIM + lane], pre2);
  }
  float h2  = tanhf(pre2);
  float w3l = W3[lane];
  float eip = h2 * w3l;
  #pragma unroll
  for (int o = 16; o > 0; o >>= 1) eip += __shfl_xor(eip, o, 32);
  float Ei = eip + b3[0] + elem_bias[element_map[atom]];
  if (lane == 0) {
    out_Ei[atom] = Ei;
    atomicAdd(&out_Etot[batch], Ei);
  }

  // ---------------- Phase 4: backward to dpre1 scale ------------------------
  float dpre2 = w3l * (1.0f - h2 * h2);
  float dg = 0.0f;
  for (int c = 0; c < DIM; ++c) {
    float dpc = __shfl(dpre2, c, 32);
    dg = fmaf(W2[lane * DIM + c], dpc, dg);
  }
  float dh = dg * (1.0f / MNB);                     // lane L holds dL/dh[*][L]

  // ---------------- Phase 5: per-neighbor backward (2 neighbors / lane) -----
  float df[2][4] = {{0, 0, 0, 0}, {0, 0, 0, 0}};
  const int j0 = lane, j1 = lane + 32;
  for (int k = 0; k < DIM; ++k) {
    float dhk = __shfl(dh, k, 32);
    float w0 = W1[0 * DIM + k], w1v = W1[1 * DIM + k],
          w2v = W1[2 * DIM + k], w3v = W1[3 * DIM + k];
    float h0 = hS[j0 * HSTR + k];
    float h1 = hS[j1 * HSTR + k];
    float dp0 = dhk * (1.0f - h0 * h0);
    float dp1 = dhk * (1.0f - h1 * h1);
    df[0][0] = fmaf(w0, dp0, df[0][0]);  df[0][1] = fmaf(w1v, dp0, df[0][1]);
    df[0][2] = fmaf(w2v, dp0, df[0][2]); df[0][3] = fmaf(w3v, dp0, df[0][3]);
    df[1][0] = fmaf(w0, dp1, df[1][0]);  df[1][1] = fmaf(w1v, dp1, df[1][1]);
    df[1][2] = fmaf(w2v, dp1, df[1][2]); df[1][3] = fmaf(w3v, dp1, df[1][3]);
  }

  float fx = 0.f, fy = 0.f, fz = 0.f;
  float vir[9] = {0, 0, 0, 0, 0, 0, 0, 0, 0};
  #pragma unroll
  for (int rep = 0; rep < 2; ++rep) {
    int j = lane + rep * 32;
    const float* row = nbS + j * NBS;
    float s = row[0], rx = row[4], ry = row[5], rz = row[6], d = row[7], sp = row[8];
    float d0 = df[rep][0], d1 = df[rep][1], d2 = df[rep][2], d3 = df[rep][3];
    float invd   = (d > 0.0f) ? 1.0f / d : 0.0f;
    float common = sp * invd * (d0 + d1 * rx + d2 * ry + d3 * rz);
    float gx = fmaf(common, rx, s * d1);
    float gy = fmaf(common, ry, s * d2);
    float gz = fmaf(common, rz, s * d3);
    fx += gx; fy += gy; fz += gz;
    float p[9] = {rx * gx, rx * gy, rx * gz,
                  ry * gx, ry * gy, ry * gz,
                  rz * gx, rz * gy, rz * gz};
    #pragma unroll
    for (int e = 0; e < 9; ++e) vir[e] += p[e];
    int idx = neighbor_indices[base + j];
    if (idx < NATOM) {                              // only real atoms are output
      float* fp = out_force + ((long)batch * NATOM + idx) * 3;
      atomicAdd(fp + 0, -gx); atomicAdd(fp + 1, -gy); atomicAdd(fp + 2, -gz);
      float* ap = out_av + ((long)batch * NATOM + idx) * 9;
      #pragma unroll
      for (int e = 0; e < 9; ++e) atomicAdd(ap + e, p[e]);
    }
  }

  // wave-reduce own-atom force and virial contributions
  #pragma unroll
  for (int o = 16; o > 0; o >>= 1) {
    fx += __shfl_xor(fx, o, 32);
    fy += __shfl_xor(fy, o, 32);
    fz += __shfl_xor(fz, o, 32);
    #pragma unroll
    for (int e = 0; e < 9; ++e) vir[e] += __shfl_xor(vir[e], o, 32);
  }
  if (lane == 0) {
    float* fp = out_force + (long)atom * 3;
    atomicAdd(fp + 0, fx); atomicAdd(fp + 1, fy); atomicAdd(fp + 2, fz);
    float* vp = out_virial + batch * 9;
    #pragma unroll
    for (int e = 0; e < 9; ++e) atomicAdd(vp + e, vir[e]);
  }
}

extern "C" void kernel_launch(void* const* d_in, const int* in_sizes, int n_in,
                              void* d_out, int out_size, void* d_ws, size_t ws_size,
                              hipStream_t stream)
{
  (void)in_sizes; (void)n_in; (void)d_ws; (void)ws_size;
  // Outputs are accumulated with atomics -> zero-init (poison is 0xAA).
  hipMemsetAsync(d_out, 0, (size_t)out_size * sizeof(float), stream);

  float* out   = (float*)d_out;
  float* Etot  = out;                                   // [2]
  float* Ei    = out + BATCH;                           // [b*n]
  float* force = Ei + (long)BATCH * NATOM;              // [b*n*3]
  float* vir   = force + (long)BATCH * NATOM * 3;       // [b*9]
  float* av    = vir + BATCH * 9;                       // [b*n*9]

  dim3 grid(BATCH * NATOM / WPB), block(WPB * 32);
  hipLaunchKernelGGL(field_fused_kernel, grid, block, 0, stream,
      (const int*)d_in[0],            // element_map
      (const int*)d_in[2],            // neighbor_indices
      (const int*)d_in[3],            // neighbor_types
      (const float*)d_in[4],          // neighbor_vectors
      (const float*)d_in[6],          // type_embed
      (const float*)d_in[7],          // elem_bias
      (const float*)d_in[8],  (const float*)d_in[9],    // W1, b1
      (const float*)d_in[10], (const float*)d_in[11],   // W2, b2
      (const float*)d_in[12], (const float*)d_in[13],   // W3, b3
      Etot, Ei, force, vir, av);
}